// multi_head_att_14370960572467
// MI455X (gfx1250) — compile-verified
//
#include <hip/hip_runtime.h>
#include <math.h>

#define B_   2
#define LF_  2048
#define LT_  2048
#define NH_  8
#define DD_  32
#define HID_ 256

typedef __attribute__((ext_vector_type(2))) float v2f;
typedef __attribute__((ext_vector_type(8))) float v8f;

// ---------------------------------------------------------------------------
// Prepass: fold Wk/Wq/Wv/Wo into per-head constants.
//   M_h = Wk_hᵀ Wq_h (3x3), u_h = Wk_hᵀ Wq_b_h, w_h = Wq_hᵀ Wk_b_h,
//   ch = Wk_b_h·Wq_b_h, c1_h = Σ_d Wv_w·Wo_w, c0_h = Σ_d Wv_b·Wo_w
// Layout per head (stride 18 floats): M[9], u[3], w[3], ch, c1, c0
// ---------------------------------------------------------------------------
__global__ __launch_bounds__(32) void mha_prep_consts(
    const float* __restrict__ Wk_w, const float* __restrict__ Wk_b,
    const float* __restrict__ Wq_w, const float* __restrict__ Wq_b,
    const float* __restrict__ Wv_w, const float* __restrict__ Wv_b,
    const float* __restrict__ Wo_w, float* __restrict__ consts)
{
    int h = threadIdx.x;
    if (h >= NH_) return;
    float M00=0,M01=0,M02=0,M10=0,M11=0,M12=0,M20=0,M21=0,M22=0;
    float u0=0,u1=0,u2=0,w0=0,w1=0,w2=0,ch=0,c1=0,c0=0;
    for (int d = 0; d < DD_; ++d) {
        int col = d * NH_ + h;                 // hidden index = d*NH + h (d-major)
        float kb = Wk_b[col], qb = Wq_b[col], wo = Wo_w[col];
        float k0 = Wk_w[0*HID_+col], k1 = Wk_w[1*HID_+col], k2 = Wk_w[2*HID_+col];
        float q0 = Wq_w[0*HID_+col], q1 = Wq_w[1*HID_+col], q2 = Wq_w[2*HID_+col];
        M00 = fmaf(k0,q0,M00); M01 = fmaf(k0,q1,M01); M02 = fmaf(k0,q2,M02);
        M10 = fmaf(k1,q0,M10); M11 = fmaf(k1,q1,M11); M12 = fmaf(k1,q2,M12);
        M20 = fmaf(k2,q0,M20); M21 = fmaf(k2,q1,M21); M22 = fmaf(k2,q2,M22);
        u0 = fmaf(k0,qb,u0); u1 = fmaf(k1,qb,u1); u2 = fmaf(k2,qb,u2);
        w0 = fmaf(kb,q0,w0); w1 = fmaf(kb,q1,w1); w2 = fmaf(kb,q2,w2);
        ch = fmaf(kb,qb,ch);
        c1 = fmaf(Wv_w[col],wo,c1);
        c0 = fmaf(Wv_b[col],wo,c0);
    }
    float* o = consts + h * 18;
    o[0]=M00; o[1]=M01; o[2]=M02; o[3]=M10; o[4]=M11; o[5]=M12;
    o[6]=M20; o[7]=M21; o[8]=M22;
    o[9]=u0;  o[10]=u1; o[11]=u2;
    o[12]=w0; o[13]=w1; o[14]=w2;
    o[15]=ch; o[16]=c1; o[17]=c0;
}

// ---------------------------------------------------------------------------
// Main kernel. One wave handles one (b, target-pair).
// WMMA 16x16x4 f32: A rows = (t0/t1 x 8 heads) of q-hat (K=4 homogeneous,
// pre-scaled by log2(e)/sqrt(HID) so scores come out in log2 units),
// B cols = 16 source points x-hat. Resulting D: each lane owns the 8 head
// base-2 scores of one (t, i) pair in its accumulators -> in-lane softmax
// via bare v_exp_f32 (exp2), no per-element log2e multiply.
// ---------------------------------------------------------------------------
__global__ __launch_bounds__(256) void mha_main(
    const float* __restrict__ coords_f, const float* __restrict__ values_f,
    const float* __restrict__ coords_t, const float* __restrict__ consts,
    const float* __restrict__ Wo_b, float* __restrict__ out)
{
    const float SCALE = 1.44269504088896340736f / 16.0f;  // log2(e)/sqrt(HID)
    int lane = threadIdx.x & 31;
    int wave = threadIdx.x >> 5;
    int task = blockIdx.x * 8 + wave;          // 0 .. B*(LT/2)-1 (wave-uniform)
    if (task >= B_ * (LT_ / 2)) return;
    int b  = task / (LT_ / 2);
    int t0 = (task % (LT_ / 2)) * 2;

    // ---- Build A-tile (q-hat). Row m: t = t0 + (m>>3), h = m&7.
    // A f32 16x4 layout: lanes 0-15 hold K=0,1 of row=lane; lanes 16-31 hold K=2,3.
    int m  = lane & 15;
    int hh = m & 7;
    int tt = t0 + (m >> 3);
    const float* yt = coords_t + ((size_t)b * LT_ + tt) * 3;
    float y0 = yt[0], y1 = yt[1], y2 = yt[2];
    const float* C = consts + hh * 18;
    float a0, a1;
    if (lane < 16) {
        a0 = (C[0]*y0 + C[1]*y1 + C[2]*y2 + C[9])   * SCALE;   // qhat[0]
        a1 = (C[3]*y0 + C[4]*y1 + C[5]*y2 + C[10])  * SCALE;   // qhat[1]
    } else {
        a0 = (C[6]*y0 + C[7]*y1 + C[8]*y2 + C[11])  * SCALE;   // qhat[2]
        a1 = (C[12]*y0 + C[13]*y1 + C[14]*y2 + C[15])* SCALE;  // qhat[3] (bias)
    }
    v2f A; A.x = a0; A.y = a1;

    float c1v[NH_], c0v[NH_];
#pragma unroll
    for (int h = 0; h < NH_; ++h) {
        c1v[h] = consts[h * 18 + 16];
        c0v[h] = consts[h * 18 + 17];
    }

    const float* xb = coords_f + (size_t)b * LF_ * 3;
    const float* vb = values_f + (size_t)b * LF_;
    float acc0 = 0.0f, acc1 = 0.0f;

    for (int i0 = 0; i0 < LF_; i0 += 32) {
#pragma unroll
        for (int half = 0; half < 2; ++half) {
            int i = i0 + half * 16 + (lane & 15);
            // B-tile (x-hat), col = i. Lanes 0-15: K=0,1; lanes 16-31: K=2,3(=1)
            const float* x = xb + (size_t)i * 3;
            float b0, b1;
            if (lane < 16) { b0 = x[0]; b1 = x[1]; }
            else           { b0 = x[2]; b1 = 1.0f; }
            v2f Bv; Bv.x = b0; Bv.y = b1;

            v8f c = {};
            c = __builtin_amdgcn_wmma_f32_16x16x4_f32(
                    /*neg_a=*/false, A, /*neg_b=*/false, Bv,
                    /*c_mod=*/(short)0, c, /*reuse_a=*/false, /*reuse_b=*/false);

            float vf = vb[i];
            // softmax over the 8 heads (base-2; identical ratios to base-e)
            float mx = fmaxf(fmaxf(fmaxf(c[0], c[1]), fmaxf(c[2], c[3])),
                             fmaxf(fmaxf(c[4], c[5]), fmaxf(c[6], c[7])));
            float sA = 0.0f, sB = 0.0f, den = 0.0f;
#pragma unroll
            for (int h = 0; h < NH_; ++h) {
                float e = __builtin_amdgcn_exp2f(c[h] - mx);  // bare v_exp_f32
                den += e;
                sA = fmaf(e, c1v[h], sA);
                sB = fmaf(e, c0v[h], sB);
            }
            float num = fmaf(vf, sA, sB);
            if (half == 0) acc0 += __fdividef(num, den);
            else           acc1 += __fdividef(num, den);
        }
    }
    float acc = acc0 + acc1;

    // ---- reduce the 16 i-stripes within each half-wave (t0: lanes 0-15, t1: 16-31)
#pragma unroll
    for (int off = 1; off < 16; off <<= 1)
        acc += __shfl_xor(acc, off, 32);

    if ((lane & 15) == 0)
        out[(size_t)b * LT_ + t0 + (lane >> 4)] = acc + Wo_b[0];
}

// ---------------------------------------------------------------------------
extern "C" void kernel_launch(void* const* d_in, const int* in_sizes, int n_in,
                              void* d_out, int out_size, void* d_ws, size_t ws_size,
                              hipStream_t stream)
{
    const float* coords_f = (const float*)d_in[0];
    const float* values_f = (const float*)d_in[1];
    const float* coords_t = (const float*)d_in[2];
    const float* Wk_w = (const float*)d_in[3];
    const float* Wk_b = (const float*)d_in[4];
    const float* Wq_w = (const float*)d_in[5];
    const float* Wq_b = (const float*)d_in[6];
    const float* Wv_w = (const float*)d_in[7];
    const float* Wv_b = (const float*)d_in[8];
    const float* Wo_w = (const float*)d_in[9];
    const float* Wo_b = (const float*)d_in[10];
    float* out    = (float*)d_out;
    float* consts = (float*)d_ws;   // 144 floats

    mha_prep_consts<<<1, 32, 0, stream>>>(Wk_w, Wk_b, Wq_w, Wq_b,
                                          Wv_w, Wv_b, Wo_w, consts);

    int tasks = B_ * (LT_ / 2);                // 2048 wave-tasks
    mha_main<<<(tasks + 7) / 8, 256, 0, stream>>>(coords_f, values_f, coords_t,
                                                  consts, Wo_b, out);
}